// GCNEncoder_3848290697594
// MI455X (gfx1250) — compile-verified
//
#include <hip/hip_runtime.h>
#include <math.h>
#include <stdint.h>

// ---------------------------------------------------------------------------
// GCN encoder for MI455X (gfx1250, wave32).
//   layer1: h1 = x @ W1 ; agg1 = elu( scatter(norm*h1[src]) + h1/deg + b1 )
//   layer2: h2 = agg1 @ W2 ; agg2 = scatter(norm*h2[src]) + h2/deg + b2
//   out    = segment_mean(agg2, batch)   -> [64,128] fp32
// GEMMs: V_WMMA_F32_16X16X4_F32 (fp32 end-to-end, workload is memory-bound)
// A tile: GLOBAL_LOAD_ASYNC_TO_LDS_B128 double-buffered (ASYNCcnt)
// B tile: K-pair-interleaved LDS layout -> each WMMA B fragment is one b64 load
// ---------------------------------------------------------------------------

typedef float v2f __attribute__((ext_vector_type(2)));
typedef float v8f __attribute__((ext_vector_type(8)));

#define NN   50000
#define NE   800000
#define CIN  256
#define CHID 256
#define COUT 128
#define NG   64

__device__ __forceinline__ void atomAddF(float* p, float v) {
  // lowers to GLOBAL_ATOMIC_ADD_F32 (no-return) instead of a CAS loop
  unsafeAtomicAdd(p, v);
}

// ---- CDNA5 async global->LDS copy (tracked by ASYNCcnt) -------------------
// ldsOff: LDS byte offset = low 32 bits of the generic __shared__ pointer.
__device__ __forceinline__ void asyncLoadB128(const void* gaddr, uint32_t ldsOff) {
  asm volatile("global_load_async_to_lds_b128 %0, %1, off"
               :: "v"(ldsOff), "v"(gaddr)
               : "memory");
}
__device__ __forceinline__ void waitAsync0() {
  asm volatile("s_wait_asynccnt 0x0" ::: "memory");
}
__device__ __forceinline__ uint32_t ldsOffsetOf(const void* p) {
  return (uint32_t)(uintptr_t)p;
}

// ---------------------------------------------------------------------------
// Tiled fp32 WMMA GEMM:  C[M,N] = A[M,K] @ B[K,N]     (K, N multiples of 16/64)
// Block: 256 threads (8 waves). Tile: BM=128 x BN=64, BK=16, double-buffered.
// Wave w computes rows [16w,16w+16) x all 64 cols (4 accumulator tiles).
// ---------------------------------------------------------------------------
#define BM 128
#define BN 64
#define BK 16
#define ASTRIDE  20   // A pad: 80B row pitch -> 16B aligned, conflict-free b64 reads
#define BPSTRIDE 160  // B pad: (BN*2 + 32) floats per K-pair -> half-waves on disjoint banks
#define BSIZE    (8 * BPSTRIDE)   // 8 K-pairs per BK=16 stage

// B tile element (k,n) lives at BsI[(k>>1)*BPSTRIDE + n*2 + (k&1)]:
// the (kb,kb+1) fragment pair for lane n is one aligned 8-byte LDS load.
__device__ __forceinline__ void storeBInterleaved(float* BsBuf, int brow, int bcol,
                                                  float4 v) {
  float* p = &BsBuf[(brow >> 1) * BPSTRIDE + bcol * 2 + (brow & 1)];
  p[0] = v.x; p[2] = v.y; p[4] = v.z; p[6] = v.w;
}

// one BK-stage of WMMAs on a staged tile pair
__device__ __forceinline__ void mma_stage(const float* AsT, const float* BsT,
                                          v8f* acc, int wid, int half, int l15) {
  #pragma unroll
  for (int ks = 0; ks < 4; ++ks) {
    const int kb   = ks * 4 + half * 2;          // K base for this half-wave
    const int pair = ks * 2 + half;              // kb >> 1
    // A: lane=M, VGPR0/1 = K kb,kb+1 (adjacent in LDS -> single b64)
    v2f a = *(const v2f*)&AsT[(wid * 16 + l15) * ASTRIDE + kb];
    #pragma unroll
    for (int ct = 0; ct < 4; ++ct) {
      // B: lane=N, VGPR0/1 = K kb,kb+1 (interleaved pair -> single b64)
      v2f b = *(const v2f*)&BsT[pair * BPSTRIDE + (ct * 16 + l15) * 2];
      acc[ct] = __builtin_amdgcn_wmma_f32_16x16x4_f32(
          false, a, false, b, (short)0, acc[ct], false, false);
    }
  }
}

__global__ __launch_bounds__(256) void gcn_gemm_wmma(
    const float* __restrict__ A, const float* __restrict__ B,
    float* __restrict__ C, int M, int K, int N)
{
  __shared__ float As[2][BM * ASTRIDE];   // 2 x 10.0 KB, async-filled
  __shared__ float BsI[2][BSIZE];         // 2 x  5.1 KB, interleaved K-pairs

  const int tid  = threadIdx.x;
  const int wid  = tid >> 5;              // wave id 0..7 -> 16-row stripe
  const int lane = tid & 31;
  const int half = lane >> 4;             // 0: lanes 0-15, 1: lanes 16-31
  const int l15  = lane & 15;

  const int m0 = blockIdx.x * BM;
  const int n0 = blockIdx.y * BN;

  v8f acc[4];
  #pragma unroll
  for (int ct = 0; ct < 4; ++ct)
    #pragma unroll
    for (int j = 0; j < 8; ++j) acc[ct][j] = 0.0f;

  // staging index precompute
  const int arow = tid >> 2;              // 0..63 (two passes of 64 rows)
  const int acol = (tid & 3) * 4;         // 0,4,8,12
  const int brow = tid >> 4;              // 0..15
  const int bcol = (tid & 15) * 4;        // 0..60

  if (m0 + BM <= M) {
    // ---------------- full tile: double-buffered pipeline ------------------
    // prologue: fill buffer 0 (A async, B via register swizzle)
    #pragma unroll
    for (int p = 0; p < 2; ++p) {
      const int r = arow + p * 64;
      asyncLoadB128(A + (size_t)(m0 + r) * K + acol,
                    ldsOffsetOf(&As[0][r * ASTRIDE + acol]));
    }
    {
      float4 bv = *(const float4*)(B + (size_t)brow * N + n0 + bcol);
      storeBInterleaved(BsI[0], brow, bcol, bv);
    }
    waitAsync0();
    __syncthreads();

    int buf = 0;
    for (int k0 = 0; k0 < K; k0 += BK) {
      const int  nxt      = buf ^ 1;
      const bool has_next = (k0 + BK) < K;
      float4 bv;
      if (has_next) {                     // issue next-stage loads, no wait yet
        #pragma unroll
        for (int p = 0; p < 2; ++p) {
          const int r = arow + p * 64;
          asyncLoadB128(A + (size_t)(m0 + r) * K + (k0 + BK) + acol,
                        ldsOffsetOf(&As[nxt][r * ASTRIDE + acol]));
        }
        bv = *(const float4*)(B + (size_t)(k0 + BK + brow) * N + n0 + bcol);
      }
      mma_stage(As[buf], BsI[buf], acc, wid, half, l15);   // overlap with loads
      if (has_next) storeBInterleaved(BsI[nxt], brow, bcol, bv);
      waitAsync0();
      __syncthreads();
      buf = nxt;
    }
  } else {
    // ---------------- M-tail block: synchronous staging with zero-fill ----
    for (int k0 = 0; k0 < K; k0 += BK) {
      __syncthreads();
      #pragma unroll
      for (int p = 0; p < 2; ++p) {
        const int r  = arow + p * 64;
        const int gr = m0 + r;
        float4 v = make_float4(0.f, 0.f, 0.f, 0.f);
        if (gr < M) v = *(const float4*)(A + (size_t)gr * K + k0 + acol);
        *(float4*)&As[0][r * ASTRIDE + acol] = v;
      }
      {
        float4 bv = *(const float4*)(B + (size_t)(k0 + brow) * N + n0 + bcol);
        storeBInterleaved(BsI[0], brow, bcol, bv);
      }
      __syncthreads();
      mma_stage(As[0], BsI[0], acc, wid, half, l15);
    }
  }

  // ---- epilogue: C VGPR j holds rows (j) and (j+8), lane%16 = column ----
  #pragma unroll
  for (int ct = 0; ct < 4; ++ct) {
    #pragma unroll
    for (int j = 0; j < 8; ++j) {
      const int row = m0 + wid * 16 + half * 8 + j;
      const int col = n0 + ct * 16 + l15;
      if (row < M) C[(size_t)row * N + col] = acc[ct][j];
    }
  }
}

// ---------------------------------------------------------------------------
// Graph-side kernels
// ---------------------------------------------------------------------------
__global__ void k_init_deg(float* deg, int n) {
  int i = blockIdx.x * blockDim.x + threadIdx.x;
  if (i < n) deg[i] = 1.0f;                       // self-loop
}

__global__ void k_count_deg(float* deg, const int* __restrict__ dst, int nE) {
  int e = blockIdx.x * blockDim.x + threadIdx.x;
  if (e < nE) atomAddF(&deg[dst[e]], 1.0f);
}

__global__ void k_final_deg(const float* __restrict__ deg,
                            float* __restrict__ dinv,
                            float* __restrict__ invdeg, int n) {
  int i = blockIdx.x * blockDim.x + threadIdx.x;
  if (i < n) {
    float d = deg[i];
    dinv[i]   = rsqrtf(d);
    invdeg[i] = 1.0f / d;
  }
}

// agg[i,c] = h[i,c] * invdeg[i]   (self-loop term, also zero-initializes agg)
__global__ void k_self_init(const float* __restrict__ h,
                            const float* __restrict__ invdeg,
                            float* __restrict__ agg, int n, int C) {
  size_t idx = (size_t)blockIdx.x * blockDim.x + threadIdx.x;
  if (idx < (size_t)n * C) agg[idx] = h[idx] * invdeg[idx / C];
}

// per-edge: agg[dst] += h[src] * dinv[src]*dinv[dst]   (one wave per edge)
__global__ __launch_bounds__(256) void k_edge_scatter(
    const float* __restrict__ h, float* __restrict__ agg,
    const int* __restrict__ src, const int* __restrict__ dst,
    const float* __restrict__ dinv, int nE, int C)
{
  int e = blockIdx.x * 8 + (threadIdx.x >> 5);
  if (e >= nE) return;
  int lane = threadIdx.x & 31;
  int s = src[e], d = dst[e];
  float nrm = dinv[s] * dinv[d];
  const float* hs = h + (size_t)s * C;
  float* ad = agg + (size_t)d * C;
  for (int c = lane * 4; c < C; c += 128) {       // coalesced 128b gathers
    float4 v = *(const float4*)(hs + c);
    atomAddF(ad + c + 0, v.x * nrm);
    atomAddF(ad + c + 1, v.y * nrm);
    atomAddF(ad + c + 2, v.z * nrm);
    atomAddF(ad + c + 3, v.w * nrm);
  }
}

// in-place: agg = (agg + bias) [elu applied if doElu]
__global__ void k_bias_act(float* __restrict__ agg, const float* __restrict__ bias,
                           int n, int C, int doElu) {
  size_t idx = (size_t)blockIdx.x * blockDim.x + threadIdx.x;
  if (idx < (size_t)n * C) {
    float v = agg[idx] + bias[idx % C];
    if (doElu) v = (v > 0.0f) ? v : expm1f(v);
    agg[idx] = v;
  }
}

// ---------------------------------------------------------------------------
// Pooling
// ---------------------------------------------------------------------------
__global__ void k_zero_pool(float* out, float* counts) {
  int i = blockIdx.x * blockDim.x + threadIdx.x;
  if (i < NG * COUT) out[i] = 0.0f;
  if (i < NG) counts[i] = 0.0f;
}

__global__ void k_node_counts(const int* __restrict__ batch, float* counts, int n) {
  int i = blockIdx.x * blockDim.x + threadIdx.x;
  if (i < n) atomAddF(&counts[batch[i]], 1.0f);
}

__global__ void k_pool_sum(const float* __restrict__ agg,
                           const int* __restrict__ batch,
                           float* __restrict__ out, int n) {
  size_t idx = (size_t)blockIdx.x * blockDim.x + threadIdx.x;
  if (idx < (size_t)n * COUT) {
    int node = (int)(idx / COUT);
    int c    = (int)(idx % COUT);
    atomAddF(&out[batch[node] * COUT + c], agg[idx]);
  }
}

__global__ void k_pool_div(float* __restrict__ out, const float* __restrict__ counts) {
  int i = blockIdx.x * blockDim.x + threadIdx.x;
  if (i < NG * COUT) out[i] /= fmaxf(counts[i / COUT], 1.0f);
}

// ---------------------------------------------------------------------------
// Host launch
// ---------------------------------------------------------------------------
extern "C" void kernel_launch(void* const* d_in, const int* in_sizes, int n_in,
                              void* d_out, int out_size, void* d_ws, size_t ws_size,
                              hipStream_t stream) {
  (void)in_sizes; (void)n_in; (void)out_size; (void)ws_size;

  const float* x    = (const float*)d_in[0];   // [50000,256]
  const float* W1   = (const float*)d_in[1];   // [256,256]
  const float* b1   = (const float*)d_in[2];   // [256]
  const float* W2   = (const float*)d_in[3];   // [256,128]
  const float* b2   = (const float*)d_in[4];   // [128]
  const int*   ei   = (const int*)d_in[5];     // [2,800000]
  const int*   bat  = (const int*)d_in[6];     // [50000]
  const int* src = ei;
  const int* dst = ei + NE;

  float* ws     = (float*)d_ws;
  float* h      = ws;                            // 50000*256 (h1, later h2)
  float* agg    = h   + (size_t)NN * CHID;       // 50000*256 (agg1, later agg2)
  float* deg    = agg + (size_t)NN * CHID;       // 50000
  float* dinv   = deg    + NN;                   // 50000
  float* invdeg = dinv   + NN;                   // 50000
  float* counts = invdeg + NN;                   // 64
  float* out    = (float*)d_out;                 // [64,128]

  const int T = 256;
  const int gNodes  = (NN + T - 1) / T;
  const int gEdges  = (NE + T - 1) / T;
  const int gScat   = (NE + 7) / 8;              // 8 edges/block, 1 wave/edge
  const int gElemH  = (NN * CHID + T - 1) / T;   // 12.8M elems
  const int gElemO  = (NN * COUT + T - 1) / T;   //  6.4M elems
  const int gM      = (NN + BM - 1) / BM;        // 391

  // degrees / normalization
  k_init_deg <<<gNodes, T, 0, stream>>>(deg, NN);
  k_count_deg<<<gEdges, T, 0, stream>>>(deg, dst, NE);
  k_final_deg<<<gNodes, T, 0, stream>>>(deg, dinv, invdeg, NN);

  // ---- layer 1 ----
  gcn_gemm_wmma<<<dim3(gM, CHID / BN), T, 0, stream>>>(x, W1, h, NN, CIN, CHID);
  k_self_init   <<<gElemH, T, 0, stream>>>(h, invdeg, agg, NN, CHID);
  k_edge_scatter<<<gScat,  T, 0, stream>>>(h, agg, src, dst, dinv, NE, CHID);
  k_bias_act    <<<gElemH, T, 0, stream>>>(agg, b1, NN, CHID, 1);

  // ---- layer 2 (h buffer reused for h2) ----
  gcn_gemm_wmma<<<dim3(gM, COUT / BN), T, 0, stream>>>(agg, W2, h, NN, CHID, COUT);
  k_self_init   <<<gElemO, T, 0, stream>>>(h, invdeg, agg, NN, COUT);
  k_edge_scatter<<<gScat,  T, 0, stream>>>(h, agg, src, dst, dinv, NE, COUT);
  k_bias_act    <<<gElemO, T, 0, stream>>>(agg, b2, NN, COUT, 0);

  // ---- global mean pool ----
  k_zero_pool  <<<(NG * COUT + T - 1) / T, T, 0, stream>>>(out, counts);
  k_node_counts<<<gNodes, T, 0, stream>>>(bat, counts, NN);
  k_pool_sum   <<<gElemO, T, 0, stream>>>(agg, bat, out, NN);
  k_pool_div   <<<(NG * COUT + T - 1) / T, T, 0, stream>>>(out, counts);
}